// PrototypicalHead_3375844295011
// MI455X (gfx1250) — compile-verified
//
#include <hip/hip_runtime.h>
#include <math.h>

typedef __attribute__((ext_vector_type(2))) float v2f;
typedef __attribute__((ext_vector_type(8))) float v8f;

#define DIMD 256
#define NCLS 512
#define EPSN 1e-8f
#define LDA  260   // padded LDS row stride in floats (260 % 64 == 4 banks -> conflict-free)

__device__ __forceinline__ float waveReduceSum(float v) {
#pragma unroll
  for (int off = 16; off > 0; off >>= 1) v += __shfl_xor(v, off, 32);
  return v;
}

__global__ __launch_bounds__(256) void proto_zero_kernel(float* p, int n) {
  int i = blockIdx.x * 256 + threadIdx.x;
  if (i < n) p[i] = 0.0f;
}

// One wave32 per support row: L2-normalize on the fly, atomically accumulate
// into per-class sums (L2-resident, 512KB) and counts.
__global__ __launch_bounds__(256) void proto_support_kernel(
    const float* __restrict__ s, const int* __restrict__ lab,
    float* __restrict__ sums, float* __restrict__ counts, int Ns) {
  int wave = threadIdx.x >> 5, lane = threadIdx.x & 31;
  int row = blockIdx.x * 8 + wave;
  if (row >= Ns) return;
  const float4* r4 = (const float4*)(s + (size_t)row * DIMD) + lane * 2;
  float4 a = r4[0], b = r4[1];
  float sq = a.x * a.x + a.y * a.y + a.z * a.z + a.w * a.w +
             b.x * b.x + b.y * b.y + b.z * b.z + b.w * b.w;
  sq = waveReduceSum(sq);
  float inv = 1.0f / fmaxf(sqrtf(sq), EPSN);
  int c = lab[row];
  float* dst = sums + (size_t)c * DIMD + lane * 8;
  __hip_atomic_fetch_add(dst + 0, a.x * inv, __ATOMIC_RELAXED, __HIP_MEMORY_SCOPE_AGENT);
  __hip_atomic_fetch_add(dst + 1, a.y * inv, __ATOMIC_RELAXED, __HIP_MEMORY_SCOPE_AGENT);
  __hip_atomic_fetch_add(dst + 2, a.z * inv, __ATOMIC_RELAXED, __HIP_MEMORY_SCOPE_AGENT);
  __hip_atomic_fetch_add(dst + 3, a.w * inv, __ATOMIC_RELAXED, __HIP_MEMORY_SCOPE_AGENT);
  __hip_atomic_fetch_add(dst + 4, b.x * inv, __ATOMIC_RELAXED, __HIP_MEMORY_SCOPE_AGENT);
  __hip_atomic_fetch_add(dst + 5, b.y * inv, __ATOMIC_RELAXED, __HIP_MEMORY_SCOPE_AGENT);
  __hip_atomic_fetch_add(dst + 6, b.z * inv, __ATOMIC_RELAXED, __HIP_MEMORY_SCOPE_AGENT);
  __hip_atomic_fetch_add(dst + 7, b.w * inv, __ATOMIC_RELAXED, __HIP_MEMORY_SCOPE_AGENT);
  if (lane == 0)
    __hip_atomic_fetch_add(counts + c, 1.0f, __ATOMIC_RELAXED, __HIP_MEMORY_SCOPE_AGENT);
}

// Global centroid numerator: sum class sums over classes (labels cover [0,C)).
__global__ __launch_bounds__(256) void proto_gsum_kernel(
    const float* __restrict__ sums, float* __restrict__ gsum) {
  int d = threadIdx.x;
  float acc = 0.0f;
  for (int c = 0; c < NCLS; ++c) acc += sums[(size_t)c * DIMD + d];
  gsum[d] = acc;
}

// One wave per class: shrink toward centroid, L2-normalize prototype.
__global__ __launch_bounds__(256) void proto_build_kernel(
    const float* __restrict__ sums, const float* __restrict__ counts,
    const float* __restrict__ gsum, const float* __restrict__ ps_ptr,
    float* __restrict__ protos, int Ns) {
  int wave = threadIdx.x >> 5, lane = threadIdx.x & 31;
  int c = blockIdx.x * 8 + wave;
  float ps = ps_ptr[0];
  float ab = 1.0f / (1.0f + expf(-ps));
  ab = fminf(fmaxf(ab, 0.0f), 0.4f);
  float cnt = fmaxf(counts[c], 1.0f);
  float alpha = ab / sqrtf(cnt);
  float rNs = 1.0f / (float)Ns;
  int d0 = lane * 8;
  float p[8];
  float sq = 0.0f;
#pragma unroll
  for (int j = 0; j < 8; ++j) {
    float mean = sums[(size_t)c * DIMD + d0 + j] / cnt;
    float gc = gsum[d0 + j] * rNs;
    p[j] = (1.0f - alpha) * mean + alpha * gc;
    sq += p[j] * p[j];
  }
  sq = waveReduceSum(sq);
  float inv = 1.0f / fmaxf(sqrtf(sq), EPSN);
#pragma unroll
  for (int j = 0; j < 8; ++j) protos[(size_t)c * DIMD + d0 + j] = p[j] * inv;
}

// One wave per query row: inverse L2 norm only (avoid writing normalized copy).
__global__ __launch_bounds__(256) void proto_qinv_kernel(
    const float* __restrict__ q, float* __restrict__ qinv, int Nq) {
  int wave = threadIdx.x >> 5, lane = threadIdx.x & 31;
  int row = blockIdx.x * 8 + wave;
  if (row >= Nq) return;
  const float4* r4 = (const float4*)(q + (size_t)row * DIMD) + lane * 2;
  float4 a = r4[0], b = r4[1];
  float sq = a.x * a.x + a.y * a.y + a.z * a.z + a.w * a.w +
             b.x * b.x + b.y * b.y + b.z * b.z + b.w * b.w;
  sq = waveReduceSum(sq);
  if (lane == 0) qinv[row] = 1.0f / fmaxf(sqrtf(sq), EPSN);
}

// WMMA fp32 GEMM with 2x2 register tiling per wave:
//   block = 8 waves = 128 query rows (4 m-strips x 2 n-halves).
//   Each wave computes four 16x16 tiles -> per k-step: 2 A-frags + 2 B-frags
//   feed 4 v_wmma_f32_16x16x4_f32 (1 LDS load per WMMA).
//   N processed in eight 64-column groups; B tile staged per group.
__global__ __launch_bounds__(256) void proto_gemm_kernel(
    const float* __restrict__ q, const float* __restrict__ qinv,
    const float* __restrict__ protos, const float* __restrict__ ls_ptr,
    float* __restrict__ out) {
  extern __shared__ float lds[];
  float* ldsA = lds;                 // 128 x LDA
  float* ldsB = lds + 128 * LDA;     // 64 x LDA
  const int tid = threadIdx.x;
  const int wave = tid >> 5, lane = tid & 31;
  const int rowBase = blockIdx.x * 128;
  const float scale = fminf(fmaxf(ls_ptr[0], 1.0f), 100.0f);

  // Stage normalized A strip: 128 rows x 64 float4
  for (int i = tid; i < 128 * 64; i += 256) {
    int r = i >> 6, kv = (i & 63) << 2;
    int rg = rowBase + r;
    float inv = qinv[rg];
    float4 v = *(const float4*)(q + (size_t)rg * DIMD + kv);
    float* d = ldsA + r * LDA + kv;
    d[0] = v.x * inv; d[1] = v.y * inv; d[2] = v.z * inv; d[3] = v.w * inv;
  }

  const int m   = lane & 15;           // M (A) / N (B) index within tile
  const int kb  = (lane >> 4) << 1;    // K sub-offset: 0 or 2
  const int mhi = (lane >> 4) << 3;    // C/D row offset: 0 or 8
  const int mIdx = wave & 3;           // which 32-row double-strip
  const int nIdx = wave >> 2;          // which 32-col half of the 64-col group
  const float* Ab0 = ldsA + (mIdx * 32 + m) * LDA;
  const float* Ab1 = Ab0 + 16 * LDA;
  const float* Bb0 = ldsB + (nIdx * 32 + m) * LDA;
  const float* Bb1 = Bb0 + 16 * LDA;

  for (int g = 0; g < NCLS / 64; ++g) {
    __syncthreads();  // previous group's ldsB reads done / ldsA visible
    const int colGroup = g * 64;
    for (int i = tid; i < 64 * 64; i += 256) {
      int r = i >> 6, kv = (i & 63) << 2;
      float4 v = *(const float4*)(protos + (size_t)(colGroup + r) * DIMD + kv);
      float* d = ldsB + r * LDA + kv;
      d[0] = v.x; d[1] = v.y; d[2] = v.z; d[3] = v.w;
    }
    __syncthreads();

    v8f acc00 = {}, acc01 = {}, acc10 = {}, acc11 = {};
#pragma unroll 4
    for (int kk = 0; kk < 64; ++kk) {
      int k = kk * 4 + kb;
      v2f a0 = *(const v2f*)(Ab0 + k);
      v2f a1 = *(const v2f*)(Ab1 + k);
      v2f b0 = *(const v2f*)(Bb0 + k);
      v2f b1 = *(const v2f*)(Bb1 + k);
      acc00 = __builtin_amdgcn_wmma_f32_16x16x4_f32(false, a0, false, b0, (short)0, acc00, false, false);
      acc01 = __builtin_amdgcn_wmma_f32_16x16x4_f32(false, a0, false, b1, (short)0, acc01, false, false);
      acc10 = __builtin_amdgcn_wmma_f32_16x16x4_f32(false, a1, false, b0, (short)0, acc10, false, false);
      acc11 = __builtin_amdgcn_wmma_f32_16x16x4_f32(false, a1, false, b1, (short)0, acc11, false, false);
    }

    const int rowT = rowBase + mIdx * 32;
    const int c0 = colGroup + nIdx * 32 + m;
    const int c1 = c0 + 16;
#pragma unroll
    for (int r = 0; r < 8; ++r) {
      size_t ro0 = (size_t)(rowT + r + mhi) * NCLS;
      size_t ro1 = (size_t)(rowT + 16 + r + mhi) * NCLS;
      out[ro0 + c0] = acc00[r] * scale;
      out[ro0 + c1] = acc01[r] * scale;
      out[ro1 + c0] = acc10[r] * scale;
      out[ro1 + c1] = acc11[r] * scale;
    }
  }
}

__global__ __launch_bounds__(256) void proto_arange_kernel(int* o, int n) {
  int i = blockIdx.x * 256 + threadIdx.x;
  if (i < n) o[i] = i;
}

extern "C" void kernel_launch(void* const* d_in, const int* in_sizes, int n_in,
                              void* d_out, int out_size, void* d_ws, size_t ws_size,
                              hipStream_t stream) {
  const float* sfeat = (const float*)d_in[0];
  const int* slab    = (const int*)d_in[1];
  const float* qfeat = (const float*)d_in[2];
  const float* lsc   = (const float*)d_in[3];
  const float* psh   = (const float*)d_in[4];
  const int Ns = in_sizes[1];
  const int Nq = in_sizes[2] / DIMD;

  float* ws     = (float*)d_ws;
  float* sums   = ws;                      // NCLS*DIMD
  float* counts = sums + NCLS * DIMD;      // NCLS
  float* gsum   = counts + NCLS;           // DIMD
  float* qinv   = gsum + DIMD;             // Nq
  float* protos = qinv + Nq;               // NCLS*DIMD
  float* out    = (float*)d_out;

  const int zn = NCLS * DIMD + NCLS + DIMD;
  proto_zero_kernel<<<(zn + 255) / 256, 256, 0, stream>>>(ws, zn);
  proto_support_kernel<<<(Ns + 7) / 8, 256, 0, stream>>>(sfeat, slab, sums, counts, Ns);
  proto_gsum_kernel<<<1, 256, 0, stream>>>(sums, gsum);
  proto_build_kernel<<<NCLS / 8, 256, 0, stream>>>(sums, counts, gsum, psh, protos, Ns);
  proto_qinv_kernel<<<(Nq + 7) / 8, 256, 0, stream>>>(qfeat, qinv, Nq);

  const size_t shmem = (size_t)(128 * LDA + 64 * LDA) * sizeof(float);  // ~200 KB
  hipFuncSetAttribute((const void*)proto_gemm_kernel,
                      hipFuncAttributeMaxDynamicSharedMemorySize, (int)shmem);
  proto_gemm_kernel<<<Nq / 128, 256, shmem, stream>>>(qfeat, qinv, protos, lsc, out);

  int* uo = ((int*)d_out) + (size_t)Nq * NCLS;
  proto_arange_kernel<<<(NCLS + 255) / 256, 256, 0, stream>>>(uo, NCLS);
}